// Orient_Conv_inout_9715216023786
// MI455X (gfx1250) — compile-verified
//
#include <hip/hip_runtime.h>
#include <math.h>

// Problem constants (match reference)
#define B_    4
#define CIN_  32
#define COUT_ 32
#define H_    128
#define W_    128
#define NPIX  (H_*W_)
#define OUTN  (B_*COUT_*H_*W_)

#define TILE  8           // output tile is TILE x TILE
#define LTILE 10          // padded tile (halo of 1 on each side)
#define LAREA (LTILE*LTILE)

// ---------------------------------------------------------------------------
// CDNA5 async global->LDS copy (ASYNCcnt path), guarded so we always compile.
// Builtin takes int* operands: src in addrspace(1), dst in addrspace(3).
// ---------------------------------------------------------------------------
#if __has_builtin(__builtin_amdgcn_global_load_async_to_lds_b32)
#define HAVE_ASYNC_LDS 1
typedef __attribute__((address_space(1))) int GI;   // global int
typedef __attribute__((address_space(3))) int LI;   // LDS int
__device__ __forceinline__ void g2l_async_b32(const float* g, float* l) {
  // emits global_load_async_to_lds_b32 (tracked by ASYNCcnt)
  __builtin_amdgcn_global_load_async_to_lds_b32((GI*)g, (LI*)l, 0, 0);
}
#else
#define HAVE_ASYNC_LDS 0
__device__ __forceinline__ void g2l_async_b32(const float* g, float* l) {
  *l = *g;   // fallback: plain load + ds_store
}
#endif

__device__ __forceinline__ void wait_async_zero() {
#if HAVE_ASYNC_LDS
#if __has_builtin(__builtin_amdgcn_s_wait_asynccnt)
  __builtin_amdgcn_s_wait_asynccnt(0);
#else
  asm volatile("s_wait_asynccnt 0" ::: "memory");
#endif
#endif
}

// ---------------------------------------------------------------------------
// Kernel 1: per (co,cin,tap) table: { cos(wr), sin(wr), w_rot0..w_rot3 }
// Rotation index maps (KH=KW=3):
//   o=0: w[i][j]     o=1: w[2-j][i]    o=2: w[2-i][2-j]   o=3: w[j][2-i]
// Gate for orientation o uses UNROTATED wr at (i,j) shifted by o*pi/2, which
// reduces to relu(u), relu(v), relu(-u), relu(-v) with
//   u = cw*r0 + sw*r1 ,  v = cw*r1 - sw*r0.
// ---------------------------------------------------------------------------
__global__ void build_tables_kernel(const float* __restrict__ w,
                                    const float* __restrict__ wr,
                                    float* __restrict__ tab) {
  int idx = blockIdx.x * blockDim.x + threadIdx.x;
  if (idx >= COUT_ * CIN_ * 9) return;
  int t  = idx % 9;
  int cc = idx / 9;             // co*CIN + cin
  int i = t / 3, j = t % 3;
  const float* wb = w + cc * 9; // w[co][cin][3][3]
  float wrv = wr[cc * 9 + t];
  float* o = tab + (size_t)idx * 6;
  o[0] = cosf(wrv);
  o[1] = sinf(wrv);
  o[2] = wb[i * 3 + j];
  o[3] = wb[(2 - j) * 3 + i];
  o[4] = wb[(2 - i) * 3 + (2 - j)];
  o[5] = wb[j * 3 + (2 - i)];
}

// ---------------------------------------------------------------------------
// Kernel 2: main gated-conv + orientation argmax.
// Block: 256 threads = 8 waves. tid = px + 64*grp, px in [0,64) is the pixel
// of an 8x8 tile, grp in [0,4) selects an 8-channel cout group (wave-uniform
// -> table reads become scalar s_loads through the constant cache).
// Each thread keeps 8 cout x 4 orientation accumulators in VGPRs.
// ---------------------------------------------------------------------------
__global__ __launch_bounds__(256)
void orient_conv_kernel(const float* __restrict__ f,
                        const float* __restrict__ r0,
                        const float* __restrict__ r1,
                        const float* __restrict__ tab,
                        float* __restrict__ out) {
  __shared__ float sf[CIN_ * LAREA];
  __shared__ float sa[CIN_ * LAREA];
  __shared__ float sb[CIN_ * LAREA];

  const int tid = threadIdx.x;
  const int blk = blockIdx.x;
  const int b   = blk >> 8;              // 256 tiles per batch image
  const int tr  = blk & 255;
  const int y0  = (tr >> 4) * TILE;
  const int x0  = (tr & 15) * TILE;

  // ---- stage halo'd tile of f/r0/r1 (all 32 cin) into LDS ----
  const int base_in = b * CIN_ * NPIX;
  for (int idx = tid; idx < CIN_ * LAREA; idx += 256) {
    int cin = idx / LAREA;
    int rr  = idx % LAREA;
    int ly = rr / LTILE, lx = rr % LTILE;
    int gy = y0 - 1 + ly, gx = x0 - 1 + lx;
    if ((unsigned)gy < (unsigned)H_ && (unsigned)gx < (unsigned)W_) {
      int g = base_in + cin * NPIX + gy * W_ + gx;
      g2l_async_b32(f  + g, &sf[idx]);
      g2l_async_b32(r0 + g, &sa[idx]);
      g2l_async_b32(r1 + g, &sb[idx]);
    } else {                     // zero padding (halo outside the image)
      sf[idx] = 0.f; sa[idx] = 0.f; sb[idx] = 0.f;
    }
  }
  wait_async_zero();
  __syncthreads();

  const int px  = tid & 63;
  const int py  = px >> 3;
  const int pxx = px & 7;
  // cout group: uniform within each wave -> uniform table addresses (SMEM path)
  const int grp = __builtin_amdgcn_readfirstlane(tid >> 6);
  const int co0 = grp * 8;

  float acc[8][4];
#pragma unroll
  for (int c = 0; c < 8; ++c)
#pragma unroll
    for (int o = 0; o < 4; ++o) acc[c][o] = 0.f;

  for (int cin = 0; cin < CIN_; ++cin) {
    // 3x3 window of f / r0 / r1 for this pixel & channel (shared over cout)
    float fw[9], aw[9], bw[9];
#pragma unroll
    for (int i = 0; i < 3; ++i)
#pragma unroll
      for (int j = 0; j < 3; ++j) {
        int s = (cin * LTILE + (py + i)) * LTILE + (pxx + j);
        fw[i * 3 + j] = sf[s];
        aw[i * 3 + j] = sa[s];
        bw[i * 3 + j] = sb[s];
      }

#pragma unroll
    for (int c = 0; c < 8; ++c) {
      const float* tc = tab + (size_t)((co0 + c) * CIN_ + cin) * 9 * 6;
#pragma unroll
      for (int t = 0; t < 9; ++t) {
        const float cw = tc[t * 6 + 0];
        const float sw = tc[t * 6 + 1];
        const float w0 = tc[t * 6 + 2];
        const float w1 = tc[t * 6 + 3];
        const float w2 = tc[t * 6 + 4];
        const float w3 = tc[t * 6 + 5];
        const float u  = fmaf(cw, aw[t],  sw * bw[t]);   // cos(wr-theta)-like
        const float v  = fmaf(cw, bw[t], -sw * aw[t]);
        const float fv = fw[t];
        acc[c][0] = fmaf(w0 * fmaxf( u, 0.f), fv, acc[c][0]);
        acc[c][1] = fmaf(w1 * fmaxf( v, 0.f), fv, acc[c][1]);
        acc[c][2] = fmaf(w2 * fmaxf(-u, 0.f), fv, acc[c][2]);
        acc[c][3] = fmaf(w3 * fmaxf(-v, 0.f), fv, acc[c][3]);
      }
    }
  }

  // ---- orientation argmax (first max wins, matching jnp.argmax) ----
  // cos/sin of bi*(pi/2) in fp32 are 4 known constants each: select them
  // branchlessly instead of calling libm (avoids Payne-Hanek code bloat).
  const int y = y0 + py, x = x0 + pxx;
#pragma unroll
  for (int c = 0; c < 8; ++c) {
    float best = acc[c][0]; int bi = 0;
    if (acc[c][1] > best) { best = acc[c][1]; bi = 1; }
    if (acc[c][2] > best) { best = acc[c][2]; bi = 2; }
    if (acc[c][3] > best) { best = acc[c][3]; bi = 3; }
    // cosf of float(k*pi/2), k=0..3
    float cv = (bi == 0) ? 1.0f
             : (bi == 1) ? -4.37113883e-8f
             : (bi == 2) ? -1.0f
             :             1.19248806e-8f;
    // sinf of float(k*pi/2), k=0..3
    float sv = (bi == 0) ? 0.0f
             : (bi == 1) ? 1.0f
             : (bi == 2) ? -8.74227766e-8f
             :             -1.0f;
    const int oi = ((b * COUT_ + co0 + c) * H_ + y) * W_ + x;
    out[oi]            = best;
    out[OUTN + oi]     = cv;
    out[2 * OUTN + oi] = sv;
  }
}

// ---------------------------------------------------------------------------
extern "C" void kernel_launch(void* const* d_in, const int* in_sizes, int n_in,
                              void* d_out, int out_size, void* d_ws, size_t ws_size,
                              hipStream_t stream) {
  const float* f  = (const float*)d_in[0];
  const float* r0 = (const float*)d_in[1];
  const float* r1 = (const float*)d_in[2];
  const float* w  = (const float*)d_in[3];
  const float* wr = (const float*)d_in[4];
  float* tab = (float*)d_ws;   // needs COUT*CIN*9*6 floats = 216 KiB

  const int ntab = COUT_ * CIN_ * 9;
  build_tables_kernel<<<(ntab + 255) / 256, 256, 0, stream>>>(w, wr, tab);

  const int nblk = B_ * (H_ / TILE) * (W_ / TILE);   // 1024 blocks
  orient_conv_kernel<<<nblk, 256, 0, stream>>>(f, r0, r1, tab, (float*)d_out);
}